// VectorQuantizer_67714454389127
// MI455X (gfx1250) — compile-verified
//
#include <hip/hip_runtime.h>
#include <math.h>

typedef float v2f __attribute__((ext_vector_type(2)));
typedef float v8f __attribute__((ext_vector_type(8)));

#define N_TOKENS 65536
#define NUM_EMB  4096
#define TILES    (NUM_EMB / 16)          // 256 code tiles of 16
#define BLOCK    256                     // 8 waves (wave32)
#define WAVES_PER_BLOCK (BLOCK / 32)
#define TOKENS_PER_WAVE 16
#define TOKENS_PER_BLOCK (WAVES_PER_BLOCK * TOKENS_PER_WAVE)  // 128
#define GRID     (N_TOKENS / TOKENS_PER_BLOCK)                // 512

// ---------------------------------------------------------------------------
// Zero the code-usage histogram (ws) before kernel 1.
// ---------------------------------------------------------------------------
__global__ void vq_zero_kernel(unsigned* __restrict__ counts) {
    int i = blockIdx.x * blockDim.x + threadIdx.x;
    if (i < NUM_EMB) counts[i] = 0u;
}

// ---------------------------------------------------------------------------
// Kernel 1: per-token argmin over the 4096-entry codebook via WMMA.
// Score[m][n] = ||w_m||^2 - 2 x_n . w_m   (argmin-equivalent to L2 distance)
// computed as a 16x16x4 f32 WMMA with augmented K:
//   A row m = [-2w0, -2w1, ||w||^2, 0] ; B col n = [x0, x1, 1, 0]
// Argmin: 8 independent accumulators (one per D-slot). Each slot r only sees
// indices of the form rowBase + r, so we store rowBase only (single cndmask
// source shared by all 8 slots) and reconstruct idx = rowBase + r at the end.
// ---------------------------------------------------------------------------
__global__ __launch_bounds__(BLOCK) void vq_argmin_kernel(
    const float* __restrict__ inputs,   // [N_TOKENS, 2]
    const float* __restrict__ weight,   // [NUM_EMB, 2]
    float* __restrict__ out_q,          // [N_TOKENS, 2]
    unsigned* __restrict__ counts,      // [NUM_EMB]
    float* __restrict__ partials)       // [GRID]
{
    // Per-lane A fragments, precomputed in the ISA's 16x4 f32 layout:
    //   lanes 0..15  (K=0,1): (-2w0, -2w1)   for m = t*16 + lane
    //   lanes 16..31 (K=2,3): (||w||^2, 0)   for m = t*16 + lane-16
    __shared__ v2f af[TILES * 32];      // 64 KB of the 320 KB WGP LDS
    __shared__ float red[BLOCK];

    const int tid = threadIdx.x;

    for (int m = tid; m < NUM_EMB; m += BLOCK) {
        float w0 = weight[2 * m], w1 = weight[2 * m + 1];
        int t = m >> 4, l = m & 15;
        v2f lo; lo.x = -2.0f * w0;        lo.y = -2.0f * w1;
        v2f hi; hi.x = w0 * w0 + w1 * w1; hi.y = 0.0f;
        af[t * 32 + l]      = lo;
        af[t * 32 + 16 + l] = hi;
    }
    __syncthreads();

    const int lane = tid & 31;
    const int wave = tid >> 5;
    const int tokenBase = (blockIdx.x * WAVES_PER_BLOCK + wave) * TOKENS_PER_WAVE;
    const int col   = lane & 15;
    const int token = tokenBase + col;

    const float x0 = inputs[2 * token];
    const float x1 = inputs[2 * token + 1];

    // B fragment (4x16 f32): K={0,1} in lanes 0..15, K={2,3} in lanes 16..31.
    v2f b;
    b.x = (lane < 16) ? x0 : 1.0f;
    b.y = (lane < 16) ? x1 : 0.0f;

    // 8 independent argmin accumulators; mb[r] holds the winning rowBase.
    float mv[8];
    int   mb[8];
#pragma unroll
    for (int r = 0; r < 8; ++r) { mv[r] = 3.0e38f; mb[r] = 0; }

    const int halfOff = (lane >= 16) ? 8 : 0;   // C/D layout: upper lanes hold rows 8..15

#pragma unroll 4
    for (int t = 0; t < TILES; ++t) {
        v2f a = af[t * 32 + lane];              // one ds_load_b64, conflict-free
        v8f c = {};
        v8f d = __builtin_amdgcn_wmma_f32_16x16x4_f32(
            /*neg_a=*/false, a, /*neg_b=*/false, b,
            /*c_mod=*/(short)0, c, /*reuse_a=*/false, /*reuse_b=*/false);
        const int rowBase = (t << 4) + halfOff;
#pragma unroll
        for (int r = 0; r < 8; ++r) {
            float v = d[r];
            // strict < keeps the first (= smallest-index) min of this residue class;
            // all 8 slots share the same rowBase cndmask source.
            if (v < mv[r]) { mv[r] = v; mb[r] = rowBase; }
        }
    }

    // Merge the 8 accumulators, reconstructing idx = rowBase + r (tie -> smaller idx).
    float minVal = mv[0];
    int   minIdx = mb[0];
#pragma unroll
    for (int r = 1; r < 8; ++r) {
        int idx = mb[r] + r;
        bool take = (mv[r] < minVal) || (mv[r] == minVal && idx < minIdx);
        if (take) { minVal = mv[r]; minIdx = idx; }
    }

    // Merge the two row-halves (lane L holds rows {..+0..7}, lane L^16 rows {..+8..15}).
    float oVal = __shfl_xor(minVal, 16, 32);
    int   oIdx = __shfl_xor(minIdx, 16, 32);
    if (oVal < minVal || (oVal == minVal && oIdx < minIdx)) {
        minVal = oVal; minIdx = oIdx;
    }

    float contrib = 0.0f;
    if (lane < 16) {
        float q0 = weight[2 * minIdx];
        float q1 = weight[2 * minIdx + 1];
        out_q[2 * token]     = q0;              // quantized_st == weight[idx] numerically
        out_q[2 * token + 1] = q1;
        float dx = q0 - x0, dy = q1 - x1;
        contrib = dx * dx + dy * dy;
        atomicAdd(&counts[minIdx], 1u);         // integer atomics: deterministic
    }

    // Deterministic per-block loss reduction.
    red[tid] = contrib;
    __syncthreads();
    for (int s = BLOCK / 2; s > 0; s >>= 1) {
        if (tid < s) red[tid] += red[tid + s];
        __syncthreads();
    }
    if (tid == 0) partials[blockIdx.x] = red[0];
}

// ---------------------------------------------------------------------------
// Kernel 2: perplexity from histogram + final q_latent_loss (single block).
// ---------------------------------------------------------------------------
__global__ __launch_bounds__(256) void vq_finalize_kernel(
    const unsigned* __restrict__ counts,
    const float* __restrict__ partials,
    float* __restrict__ out)
{
    __shared__ float sEnt[256];
    __shared__ float sLoss[256];
    const int tid = threadIdx.x;

    float ent = 0.0f;
    for (int i = tid; i < NUM_EMB; i += 256) {
        float p = (float)counts[i] * (1.0f / (float)N_TOKENS);
        ent += p * logf(p + 1e-10f);
    }
    float loss = 0.0f;
    for (int i = tid; i < GRID; i += 256) loss += partials[i];

    sEnt[tid]  = ent;
    sLoss[tid] = loss;
    __syncthreads();
    for (int s = 128; s > 0; s >>= 1) {
        if (tid < s) { sEnt[tid] += sEnt[tid + s]; sLoss[tid] += sLoss[tid + s]; }
        __syncthreads();
    }
    if (tid == 0) {
        out[N_TOKENS * 2]     = expf(-sEnt[0]);                         // perplexity
        out[N_TOKENS * 2 + 1] = sLoss[0] / (float)(N_TOKENS * 2);       // q_latent_loss
    }
}

// ---------------------------------------------------------------------------
extern "C" void kernel_launch(void* const* d_in, const int* in_sizes, int n_in,
                              void* d_out, int out_size, void* d_ws, size_t ws_size,
                              hipStream_t stream) {
    (void)in_sizes; (void)n_in; (void)out_size; (void)ws_size;

    const float* inputs = (const float*)d_in[0];   // [65536, 2] f32
    const float* weight = (const float*)d_in[1];   // [4096, 2] f32
    float* out = (float*)d_out;                    // [131072 quantized | perplexity | loss]

    // Workspace layout: [0, 4096) u32 counts ; [4096, 4608) f32 block partials
    unsigned* counts  = (unsigned*)d_ws;
    float*   partials = (float*)((char*)d_ws + NUM_EMB * sizeof(unsigned));

    vq_zero_kernel<<<(NUM_EMB + 255) / 256, 256, 0, stream>>>(counts);
    vq_argmin_kernel<<<GRID, BLOCK, 0, stream>>>(inputs, weight, out, counts, partials);
    vq_finalize_kernel<<<1, 256, 0, stream>>>(counts, partials, out);
}